// FraudDetectionModel_65481071405351
// MI455X (gfx1250) — compile-verified
//
#include <hip/hip_runtime.h>
#include <math.h>

// CDNA5 / gfx1250: wave32, WMMA 16x16x4 f32, 320KB LDS per WGP.
// Whole circuit runs in ONE workgroup, fully LDS-resident.
// Inputs (131 floats) are staged global->LDS once via ASYNC load when available.

typedef __attribute__((ext_vector_type(2))) float v2f;
typedef __attribute__((ext_vector_type(8))) float v8f;

#define CUTN     10
#define NLAYERS  8
#define BIGD     100     // valid dim of BS matrix
#define PD       112     // padded dim (7 tiles of 16)
#define NTILE    7
#define NTAYLOR  12
#define BLOCK    256     // 8 waves

#define PLANE    (PD*PD)               // 12544 floats
#define NPARAM   131                   // x(2) + params(126) + W(2) + b(1)
#define LDS_FLOATS (6*PLANE + 2*256 /*psi*/ + 2*256 /*T*/ + 2*128 /*vec*/ + 2*256 /*red*/ + 132 /*pbuf*/)

#if defined(__gfx1250__) && __has_builtin(__builtin_amdgcn_global_load_async_to_lds_b32) && __has_builtin(__builtin_amdgcn_s_wait_asynccnt)
#define HAVE_ASYNC_LDS 1
typedef __attribute__((address_space(1))) int gint_t;   // global (AS1)
typedef __attribute__((address_space(3))) int lint_t;   // LDS    (AS3)
#else
#define HAVE_ASYNC_LDS 0
#endif

__device__ __forceinline__ v8f wmma_f32(v2f a, v2f b, v8f c) {
  // V_WMMA_F32_16X16X4_F32 : D = A(16x4) * B(4x16) + C(16x16)
  return __builtin_amdgcn_wmma_f32_16x16x4_f32(
      /*neg_a=*/false, a, /*neg_b=*/false, b,
      /*c_mod=*/(short)0, c, /*reuse_a=*/false, /*reuse_b=*/false);
}

__device__ __forceinline__ v8f zero8() {
  v8f z = {0.f,0.f,0.f,0.f,0.f,0.f,0.f,0.f};
  return z;
}
__device__ __forceinline__ float clip5(float v){ return fminf(fmaxf(v,-5.f),5.f); }
__device__ __forceinline__ float clip1(float v){ return fminf(fmaxf(v,-1.f),1.f); }

// ---------------- big 112x112 complex matmul, D = scale*(A@B) (+I), all 8 waves ------------
__device__ void big_cmm(const float* Ar, const float* Ai,
                        const float* Br, const float* Bi,
                        float* Dr, float* Di,
                        int wave, int lane, float scale, int addI)
{
  const int lhi = lane >> 4, lln = lane & 15;
  for (int t = wave; t < NTILE*NTILE; t += 8) {
    const int row0 = (t / NTILE) * 16;
    const int col0 = (t % NTILE) * 16;
    v8f rr = zero8(), ii = zero8(), ri = zero8(), ir = zero8();
    for (int k0 = 0; k0 < BIGD; k0 += 4) {           // K needs no padding: 100 % 4 == 0
      const int ka   = k0 + 2*lhi;
      const int aoff = (row0 + lln) * PD + ka;
      v2f a_r; a_r.x = Ar[aoff]; a_r.y = Ar[aoff + 1];
      v2f a_i; a_i.x = Ai[aoff]; a_i.y = Ai[aoff + 1];
      const int boff = ka * PD + col0 + lln;
      v2f b_r; b_r.x = Br[boff]; b_r.y = Br[boff + PD];
      v2f b_i; b_i.x = Bi[boff]; b_i.y = Bi[boff + PD];
      rr = wmma_f32(a_r, b_r, rr);
      ii = wmma_f32(a_i, b_i, ii);
      ri = wmma_f32(a_r, b_i, ri);
      ir = wmma_f32(a_i, b_r, ir);
    }
#pragma unroll
    for (int v = 0; v < 8; ++v) {
      const int m = row0 + v + 8*lhi;
      const int n = col0 + lln;
      float dr = (rr[v] - ii[v]) * scale;
      float di = (ri[v] + ir[v]) * scale;
      if (addI && m == n && m < BIGD) dr += 1.0f;
      Dr[m*PD + n] = dr;
      Di[m*PD + n] = di;
    }
  }
}

// ---------------- build BS Hamiltonian (scaled): H = tscl*(e kron(AD,A) - conj(e) kron(A,AD)) --
__device__ void build_H(float* Hr, float* Hi, float bphi, float tscl, int tid)
{
  const float cph = cosf(bphi), sph = sinf(bphi);
  for (int idx = tid; idx < PLANE; idx += BLOCK) {
    const int r = idx / PD, c = idx % PD;
    float hr = 0.f, hi = 0.f;
    if (r < BIGD && c < BIGD) {
      const int i = r / CUTN, j = r % CUTN;
      const int k = c / CUTN, l = c % CUTN;
      if (k == i-1 && l == j+1) {                 // e * sqrt(i)*sqrt(j+1)
        const float w = sqrtf((float)(i*(j+1)));
        hr += cph * w; hi += sph * w;
      }
      if (k == i+1 && l == j-1) {                 // -conj(e) * sqrt(i+1)*sqrt(j)
        const float w = sqrtf((float)((i+1)*j));
        hr -= cph * w; hi += sph * w;
      }
    }
    Hr[idx] = hr * tscl;
    Hi[idx] = hi * tscl;
  }
}

// ---------------- big expm via Horner-Taylor + squaring; result always ends in B0 -------------
__device__ void big_expm(float* Hr, float* Hi,
                         float* B0r, float* B0i, float* B1r, float* B1i,
                         int s, int tid, int wave, int lane)
{
  const int T = NTAYLOR + s;
  float *Pr, *Pi, *Qr, *Qi;
  if (T & 1) { Pr = B1r; Pi = B1i; Qr = B0r; Qi = B0i; }
  else       { Pr = B0r; Pi = B0i; Qr = B1r; Qi = B1i; }
  for (int idx = tid; idx < PLANE; idx += BLOCK) {   // P = I (valid block only)
    const int r = idx / PD, c = idx % PD;
    Pr[idx] = (r == c && r < BIGD) ? 1.f : 0.f;
    Pi[idx] = 0.f;
  }
  __syncthreads();
  for (int k = NTAYLOR; k >= 1; --k) {               // P <- I + (H@P)/k
    big_cmm(Hr, Hi, Pr, Pi, Qr, Qi, wave, lane, 1.0f/(float)k, 1);
    __syncthreads();
    float* t; t=Pr;Pr=Qr;Qr=t; t=Pi;Pi=Qi;Qi=t;
  }
  for (int q = 0; q < s; ++q) {                      // P <- P@P
    big_cmm(Pr, Pi, Pr, Pi, Qr, Qi, wave, lane, 1.0f, 0);
    __syncthreads();
    float* t; t=Pr;Pr=Qr;Qr=t; t=Pi;Pi=Qi;Qi=t;
  }
}

// ---------------- single-tile 16x16 complex matmul (one wave), optional B-transpose ----------
__device__ void cmm16(const float* Ar, const float* Ai,
                      const float* Br, const float* Bi,
                      float* Dr, float* Di,
                      int lane, float scale, int addI, int transB)
{
  const int lhi = lane >> 4, lln = lane & 15;
  v8f rr = zero8(), ii = zero8(), ri = zero8(), ir = zero8();
#pragma unroll
  for (int k0 = 0; k0 < 16; k0 += 4) {
    const int ka = k0 + 2*lhi;
    v2f a_r, a_i, b_r, b_i;
    a_r.x = Ar[lln*16 + ka]; a_r.y = Ar[lln*16 + ka + 1];
    a_i.x = Ai[lln*16 + ka]; a_i.y = Ai[lln*16 + ka + 1];
    if (transB) {       // use B[n][k] -> computes A @ B^T
      b_r.x = Br[lln*16 + ka];      b_r.y = Br[lln*16 + ka + 1];
      b_i.x = Bi[lln*16 + ka];      b_i.y = Bi[lln*16 + ka + 1];
    } else {
      b_r.x = Br[ka*16 + lln];      b_r.y = Br[(ka+1)*16 + lln];
      b_i.x = Bi[ka*16 + lln];      b_i.y = Bi[(ka+1)*16 + lln];
    }
    rr = wmma_f32(a_r, b_r, rr);
    ii = wmma_f32(a_i, b_i, ii);
    ri = wmma_f32(a_r, b_i, ri);
    ir = wmma_f32(a_i, b_r, ir);
  }
#pragma unroll
  for (int v = 0; v < 8; ++v) {
    const int m = v + 8*lhi, n = lln;
    float dr = (rr[v] - ii[v]) * scale;
    float di = (ri[v] + ir[v]) * scale;
    if (addI && m == n && m < CUTN) dr += 1.0f;
    Dr[m*16 + n] = dr;
    Di[m*16 + n] = di;
  }
}

// ---------------- small 10x10 expm in a 16x16 padded tile; one wave; result -> arena[0..511] --
__device__ void small_expm(float* arena, int lane, int isSqueeze, float r, float phi)
{
  float* Mr  = arena;        float* Mi  = arena + 256;
  float* P0r = arena + 512;  float* P0i = arena + 768;
  float* P1r = arena + 1024; float* P1i = arena + 1280;

  const float cph = cosf(phi), sph = sinf(phi);
  float bound = (isSqueeze ? 9.0f : 6.5f) * fabsf(r);   // inf-norm bound of generator
  int s = 0;
  while (bound > 0.25f && s < 24) { bound *= 0.5f; s++; }
  const float scl = 1.0f / (float)(1 << s);

  for (int e = lane; e < 256; e += 32) {
    const int p = e >> 4, q = e & 15;
    float mr = 0.f, mi = 0.f;
    if (p < CUTN && q < CUTN) {
      if (isSqueeze) {      // 0.5*(conj(z)A^2 - z AD^2), z = r e^{i phi}
        if (q == p + 2) {
          const float w = 0.5f * r * sqrtf((float)((p+1)*(p+2)));
          mr += cph * w; mi -= sph * w;
        }
        if (q == p - 2) {
          const float w = 0.5f * r * sqrtf((float)(p*(p-1)));
          mr -= cph * w; mi -= sph * w;
        }
      } else {              // alpha*AD - conj(alpha)*A, alpha = r e^{i phi}
        if (q == p - 1) {
          const float w = r * sqrtf((float)p);
          mr += cph * w; mi += sph * w;
        }
        if (q == p + 1) {
          const float w = r * sqrtf((float)(p+1));
          mr -= cph * w; mi += sph * w;
        }
      }
    }
    Mr[e] = mr * scl;
    Mi[e] = mi * scl;
  }

  const int T = NTAYLOR + s;
  float *Pr, *Pi, *Qr, *Qi;
  if (T & 1) { Pr=P1r; Pi=P1i; Qr=P0r; Qi=P0i; } else { Pr=P0r; Pi=P0i; Qr=P1r; Qi=P1i; }
  for (int e = lane; e < 256; e += 32) {
    const int p = e >> 4, q = e & 15;
    Pr[e] = (p == q && p < CUTN) ? 1.f : 0.f;
    Pi[e] = 0.f;
  }
  for (int k = NTAYLOR; k >= 1; --k) {
    cmm16(Mr, Mi, Pr, Pi, Qr, Qi, lane, 1.0f/(float)k, 1, 0);
    float* t; t=Pr;Pr=Qr;Qr=t; t=Pi;Pi=Qi;Qi=t;
  }
  for (int q2 = 0; q2 < s; ++q2) {
    cmm16(Pr, Pi, Pr, Pi, Qr, Qi, lane, 1.0f, 0, 0);
    float* t; t=Pr;Pr=Qr;Qr=t; t=Pi;Pi=Qi;Qi=t;
  }
  for (int e = lane; e < 256; e += 32) { Mr[e] = Pr[e]; Mi[e] = Pi[e]; }  // fixed location
}

// ---------------- psi <- L @ psi @ R^T  (wave 0 only) ---------------------------------------
__device__ void apply_pair(const float* Lr, const float* Li,
                           const float* Rr, const float* Ri,
                           float* psir, float* psii, float* tr, float* ti, int lane)
{
  cmm16(Lr, Li, psir, psii, tr, ti, lane, 1.0f, 0, 0);
  cmm16(tr, ti, Rr, Ri, psir, psii, lane, 1.0f, 0, 1);
}

// ---------------- psi_vec <- BS @ vec(psi)  (100-wide VALU matvec, all threads) --------------
__device__ void bs_matvec(const float* Br, const float* Bi,
                          float* psir, float* psii, float* vr, float* vi, int tid)
{
  __syncthreads();
  if (tid < BIGD) {
    float sr = 0.f, si = 0.f;
    for (int c = 0; c < BIGD; ++c) {
      const int e = (c / CUTN) * 16 + (c % CUTN);
      const float pr = psir[e], pi = psii[e];
      const float br = Br[tid*PD + c], bi = Bi[tid*PD + c];
      sr += br*pr - bi*pi;
      si += br*pi + bi*pr;
    }
    vr[tid] = sr; vi[tid] = si;
  }
  __syncthreads();
  if (tid < BIGD) {
    const int e = (tid / CUTN) * 16 + (tid % CUTN);
    psir[e] = vr[tid];
    psii[e] = vi[tid];
  }
  __syncthreads();
}

__device__ void phase_mul(float* psir, float* psii, float a0, float a1, int tid, int squared)
{
  if (tid < BIGD) {
    const int i = tid / CUTN, j = tid % CUTN;
    const float ang = squared ? (a0*(float)(i*i) + a1*(float)(j*j))
                              : (a0*(float)i + a1*(float)j);
    const float c = cosf(ang), s = sinf(ang);
    const int e = i*16 + j;
    const float pr = psir[e], pi = psii[e];
    psir[e] = pr*c - pi*s;
    psii[e] = pr*s + pi*c;
  }
}

// =============================================================================================
__global__ void qphot_circuit_kernel(const float* __restrict__ x,
                                     const float* __restrict__ params,
                                     const float* __restrict__ W,
                                     const float* __restrict__ bvec,
                                     float* __restrict__ out)
{
  extern __shared__ float sm[];
  const int tid  = threadIdx.x;
  const int wave = tid >> 5;
  const int lane = tid & 31;

  float* Hr   = sm;                 // also hosts 4 small-expm arenas (1536 floats each)
  float* Hi   = sm + PLANE;
  float* B0r  = sm + 2*PLANE;
  float* B0i  = sm + 3*PLANE;
  float* B1r  = sm + 4*PLANE;
  float* B1i  = sm + 5*PLANE;
  float* psir = sm + 6*PLANE;
  float* psii = psir + 256;
  float* tr   = psii + 256;
  float* ti   = tr   + 256;
  float* vr   = ti   + 256;
  float* vi   = vr   + 128;
  float* red0 = vi   + 128;
  float* red1 = red0 + 256;
  float* pbuf = red1 + 256;         // staged inputs: x[0..1], params[2..127], W[128..129], b[130]

  // ---- stage all inputs global -> LDS once (async path on CDNA5) ----
#if HAVE_ASYNC_LDS
  {
    const float* src = (tid < 2)   ? (x + tid)
                     : (tid < 128) ? (params + (tid - 2))
                     : (tid < 130) ? (W + (tid - 128))
                     :               bvec;
    if (tid < NPARAM) {
      __builtin_amdgcn_global_load_async_to_lds_b32(
          (gint_t*)src,
          (lint_t*)(pbuf + tid),
          0, 0);
    }
    __builtin_amdgcn_s_wait_asynccnt(0);
  }
#else
  if (tid == 0) __builtin_prefetch(params, 0, 3);
  if (tid < 2)         pbuf[tid] = x[tid];
  else if (tid < 128)  pbuf[tid] = params[tid - 2];
  else if (tid < 130)  pbuf[tid] = W[tid - 128];
  else if (tid == 130) pbuf[tid] = bvec[0];
#endif

  // psi = |0,0>
  for (int e = tid; e < 256; e += BLOCK) { psir[e] = 0.f; psii[e] = 0.f; }
  if (tid == 0) psir[0] = 1.0f;
  __syncthreads();

  // initial displacement by x (phi = 0); arenas live in the (currently unused) H region
  if      (wave == 0) small_expm(Hr + 0*1536, lane, 0, pbuf[0], 0.0f);
  else if (wave == 1) small_expm(Hr + 1*1536, lane, 0, pbuf[1], 0.0f);
  __syncthreads();
  if (wave == 0) apply_pair(Hr,       Hr+256,
                            Hr+1536,  Hr+1536+256, psir, psii, tr, ti, lane);
  __syncthreads();

  for (int li = 0; li <= NLAYERS; ++li) {
    const float* p = pbuf + 2 + li*14;
    const float theta = p[0], bphi = p[1];
    const float ph0 = p[2], ph1 = p[3];
    float sr0 = p[4], sr1 = p[5];
    const float sp0 = p[6], sp1 = p[7];
    float dr0 = p[8], dr1 = p[9];
    const float dp0 = p[10], dp1 = p[11];
    float kk0 = p[12], kk1 = p[13];
    if (li >= 1) {
      sr0 = clip5(sr0); sr1 = clip5(sr1);
      dr0 = clip5(dr0); dr1 = clip5(dr1);
      kk0 = clip1(kk0); kk1 = clip1(kk1);
    }

    // ---- BS = expm(theta * H), scaling & squaring ----
    float nb = fabsf(theta) * 19.0f;               // inf-norm bound of theta*H
    int s = 0;
    while (nb > 0.25f && s < 24) { nb *= 0.5f; s++; }
    const float tscl = theta / (float)(1 << s);
    build_H(Hr, Hi, bphi, tscl, tid);
    __syncthreads();
    big_expm(Hr, Hi, B0r, B0i, B1r, B1i, s, tid, wave, lane);   // result in B0*

    // ---- small expms: S0, S1, D0, D1 (H region is dead now; reuse as arenas) ----
    if      (wave == 0) small_expm(Hr + 0*1536, lane, 1, sr0, sp0);
    else if (wave == 1) small_expm(Hr + 1*1536, lane, 1, sr1, sp1);
    else if (wave == 2) small_expm(Hr + 2*1536, lane, 0, dr0, dp0);
    else if (wave == 3) small_expm(Hr + 3*1536, lane, 0, dr1, dp1);
    __syncthreads();

    // ---- layer application ----
    bs_matvec(B0r, B0i, psir, psii, vr, vi, tid);
    phase_mul(psir, psii, ph0, ph1, tid, 0);
    __syncthreads();
    if (wave == 0) apply_pair(Hr,           Hr+256,            // S0
                              Hr+1536,      Hr+1536+256,       // S1 (transposed)
                              psir, psii, tr, ti, lane);
    __syncthreads();
    bs_matvec(B0r, B0i, psir, psii, vr, vi, tid);
    phase_mul(psir, psii, ph0, ph1, tid, 0);
    __syncthreads();
    if (wave == 0) apply_pair(Hr+2*1536,    Hr+2*1536+256,     // D0
                              Hr+3*1536,    Hr+3*1536+256,     // D1 (transposed)
                              psir, psii, tr, ti, lane);
    __syncthreads();
    phase_mul(psir, psii, kk0, kk1, tid, 1);                    // Kerr
    __syncthreads();
  }

  // ---- readout: n0, n1, then linear head ----
  float a0 = 0.f, a1 = 0.f;
  if (tid < BIGD) {
    const int i = tid / CUTN, j = tid % CUTN;
    const int e = i*16 + j;
    const float pr = psir[e], pi = psii[e];
    const float pb = pr*pr + pi*pi;
    a0 = pb * (float)i;
    a1 = pb * (float)j;
  }
  red0[tid] = a0; red1[tid] = a1;
  __syncthreads();
  for (int off = 128; off > 0; off >>= 1) {
    if (tid < off) { red0[tid] += red0[tid+off]; red1[tid] += red1[tid+off]; }
    __syncthreads();
  }
  if (tid == 0) out[0] = red0[0]*pbuf[128] + red1[0]*pbuf[129] + pbuf[130];
}

extern "C" void kernel_launch(void* const* d_in, const int* in_sizes, int n_in,
                              void* d_out, int out_size, void* d_ws, size_t ws_size,
                              hipStream_t stream) {
  (void)in_sizes; (void)n_in; (void)d_ws; (void)ws_size; (void)out_size;
  const float* x      = (const float*)d_in[0];
  const float* params = (const float*)d_in[1];
  const float* W      = (const float*)d_in[2];
  const float* b      = (const float*)d_in[3];
  float* out = (float*)d_out;
  const size_t lds_bytes = (size_t)LDS_FLOATS * sizeof(float);  // ~309 KB <= 320 KB/WGP
  qphot_circuit_kernel<<<1, BLOCK, lds_bytes, stream>>>(x, params, W, b, out);
}